// DynamicGraphAttention_11596411699419
// MI455X (gfx1250) — compile-verified
//
#include <hip/hip_runtime.h>
#include <cstddef>

typedef float v2f __attribute__((ext_vector_type(2)));
typedef float v8f __attribute__((ext_vector_type(8)));

#define B_  256
#define N_  64
#define T_  4096
#define H_  16
#define TC  128          // T-chunk per block in k_out
#define XS  140          // padded LDS stride for x tile (2*XS%64=24, 8*XS%64=32, 16B rows)
#define AS  68           // padded LDS stride for attn tile (spreads A gathers over all banks)

// ---- CDNA5 async global->LDS DMA (GLOBAL_LOAD_ASYNC_TO_LDS_B128, ASYNCcnt) ----
// GV addressing: vdst = wave-relative LDS byte offset (low 32 bits of generic
// __shared__ pointer per the flat-aperture truncation rule), vaddr = 64-bit
// global address.
__device__ __forceinline__ void async_load_b128(const void* gsrc, const void* lds_dst) {
    asm volatile("global_load_async_to_lds_b128 %0, %1, off"
                 :: "v"((unsigned int)(size_t)lds_dst), "v"(gsrc)
                 : "memory");
}
__device__ __forceinline__ void wait_asynccnt0() {
    asm volatile("s_wait_asynccnt 0" ::: "memory");
}

// ---------------- Kernel 1: energy = var(x, axis=T) ----------------
__global__ __launch_bounds__(256) void k_energy(const float* __restrict__ x,
                                                float* __restrict__ energy) {
    const int bn = blockIdx.x;                 // 0 .. B*N-1
    const float* p = x + (size_t)bn * T_;
    const int tid = threadIdx.x;

    float s = 0.f, ss = 0.f;
#pragma unroll
    for (int k = 0; k < 4; ++k) {
        float4 v = *(const float4*)(p + 4 * (tid + 256 * k));
        s  += v.x + v.y + v.z + v.w;
        ss += v.x * v.x + v.y * v.y + v.z * v.z + v.w * v.w;
    }
#pragma unroll
    for (int off = 16; off > 0; off >>= 1) {
        s  += __shfl_xor(s,  off, 32);
        ss += __shfl_xor(ss, off, 32);
    }
    __shared__ float ls[8], lss[8];
    const int wave = tid >> 5, lane = tid & 31;
    if (lane == 0) { ls[wave] = s; lss[wave] = ss; }
    __syncthreads();
    if (tid == 0) {
        float S = 0.f, SS = 0.f;
#pragma unroll
        for (int w = 0; w < 8; ++w) { S += ls[w]; SS += lss[w]; }
        const float inv = 1.0f / (float)T_;
        const float mean = S * inv;
        energy[bn] = SS * inv - mean * mean;   // population variance (ddof=0)
    }
}

// ---------------- Kernel 2: attn_w = softmax(masked rank-1 logits) ----------------
__global__ __launch_bounds__(64) void k_attn(const float* __restrict__ energy,
                                             const float* __restrict__ Wq,
                                             const float* __restrict__ bq,
                                             const float* __restrict__ Wk,
                                             const float* __restrict__ bk,
                                             float* __restrict__ attn) {
    const int b = blockIdx.x;
    const int i = threadIdx.x;                 // row 0..63
    __shared__ float e[N_];
    e[i] = energy[b * N_ + i];

    float c1 = 0.f, c2 = 0.f, c3 = 0.f, c4 = 0.f;
#pragma unroll
    for (int h = 0; h < H_; ++h) {
        c1 += Wq[h] * Wk[h];
        c2 += Wq[h] * bk[h];
        c3 += bq[h] * Wk[h];
        c4 += bq[h] * bk[h];
    }
    __syncthreads();

    const float scale = 0.25f;                 // H^-0.5
    const float ei = e[i];
    float logit[N_];
    float m = -3.402823466e38f;
#pragma unroll
    for (int j = 0; j < N_; ++j) {
        float l = (j == i) ? -1e9f
                           : scale * (ei * e[j] * c1 + ei * c2 + e[j] * c3 + c4);
        logit[j] = l;
        m = fmaxf(m, l);
    }
    float sum = 0.f;
#pragma unroll
    for (int j = 0; j < N_; ++j) {
        float w = __expf(logit[j] - m);
        logit[j] = w;
        sum += w;
    }
    const float inv = 1.0f / sum;
    float* row = attn + ((size_t)b * N_ + i) * N_;
#pragma unroll
    for (int j = 0; j < N_; ++j) row[j] = logit[j] * inv;
}

// ---------------- Kernel 3: out = attn_w @ x + x  (WMMA f32 16x16x4) ----------------
__global__ __launch_bounds__(256) void k_out(const float* __restrict__ x,
                                             const float* __restrict__ attn,
                                             float* __restrict__ out) {
    __shared__ float xs[N_ * XS];              // 64 x 128 x-tile, padded
    __shared__ float as[N_ * AS];              // 64 x 64 attn tile, padded

    const int b  = blockIdx.x;
    const int t0 = blockIdx.y * TC;
    const int tid = threadIdx.x;

    // ---- Stage tiles with async global->LDS DMA ----
    const float* xb = x + (size_t)b * N_ * T_;
#pragma unroll
    for (int it = 0; it < 8; ++it) {           // 64x128 = 2048 float4
        int idx  = it * 256 + tid;
        int row  = idx >> 5;                   // 32 float4 per row
        int col4 = idx & 31;
        async_load_b128(xb + (size_t)row * T_ + t0 + col4 * 4,
                        &xs[row * XS + col4 * 4]);
    }
    const float* ab = attn + (size_t)b * N_ * N_;
#pragma unroll
    for (int it = 0; it < 4; ++it) {           // 64x64 = 1024 float4
        int idx  = it * 256 + tid;
        int row  = idx >> 4;                   // 16 float4 per row
        int col4 = idx & 15;
        async_load_b128(ab + (size_t)idx * 4, &as[row * AS + col4 * 4]);
    }
    wait_asynccnt0();
    __syncthreads();

    const int wave = tid >> 5;
    const int lane = tid & 31;
    const int m    = lane & 15;                // M (A rows) / N (B,C,D cols)
    const int hi   = lane >> 4;
    const int rtp  = wave & 1;                 // row pair: rows rtp*32 .. rtp*32+31
    const int cg   = wave >> 1;                // col group: colTiles {2cg, 2cg+1}

    // A operands in registers: two 16x64 attn slices (reused across col tiles).
    // ISA 32-bit A 16x4 layout: lanes 0-15 hold K=0(v0),K=1(v1); lanes 16-31 K=2,3.
    v2f a[2][16];
#pragma unroll
    for (int r = 0; r < 2; ++r) {
        const int i = (rtp * 2 + r) * 16 + m;
#pragma unroll
        for (int ks = 0; ks < 16; ++ks) {
            int base = i * AS + ks * 4 + hi * 2;
            a[r][ks] = v2f{ as[base], as[base + 1] };
        }
    }

#pragma unroll
    for (int cc = 0; cc < 2; ++cc) {
        const int tl = (cg * 2 + cc) * 16;

        // C/D layout: VGPR r -> M = r (lanes 0-15) / r+8 (lanes 16-31), N = lane&15.
        // Initialize accumulators with residual x (fuses the "+ x").
        v8f acc0, acc1;
#pragma unroll
        for (int r8 = 0; r8 < 8; ++r8) {
            acc0[r8] = xs[((rtp * 2 + 0) * 16 + r8 + hi * 8) * XS + tl + m];
            acc1[r8] = xs[((rtp * 2 + 1) * 16 + r8 + hi * 8) * XS + tl + m];
        }

        // Each B load feeds two independent WMMA chains (2x A reuse).
#pragma unroll
        for (int ks = 0; ks < 16; ++ks) {
            const int kr = ks * 4 + hi * 2;    // B: lanes 0-15 K=0,1; lanes 16-31 K=2,3
            v2f bv = v2f{ xs[kr * XS + tl + m], xs[(kr + 1) * XS + tl + m] };
            acc0 = __builtin_amdgcn_wmma_f32_16x16x4_f32(
                false, a[0][ks], false, bv, (short)0, acc0, false, false);
            acc1 = __builtin_amdgcn_wmma_f32_16x16x4_f32(
                false, a[1][ks], false, bv, (short)0, acc1, false, false);
        }

#pragma unroll
        for (int r8 = 0; r8 < 8; ++r8) {
            int row0 = (rtp * 2 + 0) * 16 + r8 + hi * 8;
            int row1 = (rtp * 2 + 1) * 16 + r8 + hi * 8;
            __builtin_nontemporal_store(
                acc0[r8], out + ((size_t)b * N_ + row0) * T_ + t0 + tl + m);
            __builtin_nontemporal_store(
                acc1[r8], out + ((size_t)b * N_ + row1) * T_ + t0 + tl + m);
        }
    }
}

extern "C" void kernel_launch(void* const* d_in, const int* in_sizes, int n_in,
                              void* d_out, int out_size, void* d_ws, size_t ws_size,
                              hipStream_t stream) {
    const float* x  = (const float*)d_in[0];
    const float* Wq = (const float*)d_in[1];
    const float* bq = (const float*)d_in[2];
    const float* Wk = (const float*)d_in[3];
    const float* bk = (const float*)d_in[4];

    float* out  = (float*)d_out;
    float* attn = out + (size_t)B_ * N_ * T_;  // second tuple output, concatenated flat
    float* energy = (float*)d_ws;              // B*N floats of scratch

    k_energy<<<B_ * N_, 256, 0, stream>>>(x, energy);
    k_attn<<<B_, 64, 0, stream>>>(energy, Wq, bq, Wk, bk, attn);
    k_out<<<dim3(B_, T_ / TC), 256, 0, stream>>>(x, attn, out);
}